// MultiHeadedCausalSelfAttention_81647328297553
// MI455X (gfx1250) — compile-verified
//
#include <hip/hip_runtime.h>
#include <math.h>

#define T_DIM 4096
#define EMBED 1024
#define HEADS 16
#define HDIM  64   // EMBED / HEADS

typedef float v2f __attribute__((ext_vector_type(2)));
typedef float v8f __attribute__((ext_vector_type(8)));
typedef int   i4v __attribute__((vector_size(4 * sizeof(int))));

// ---------------------------------------------------------------------------
// CDNA5 async global->LDS copy (ASYNCcnt-tracked). Guarded: falls back to a
// synchronous copy if the toolchain lacks the builtin.
// ---------------------------------------------------------------------------
#if __has_builtin(__builtin_amdgcn_global_load_async_to_lds_b128)
#define HAVE_ASYNC_COPY 1
__device__ __forceinline__ void copy16_g2l(const void* gsrc, void* ldst) {
  __builtin_amdgcn_global_load_async_to_lds_b128((i4v*)gsrc, (i4v*)ldst, 0, 0);
}
#else
#define HAVE_ASYNC_COPY 0
__device__ __forceinline__ void copy16_g2l(const void* gsrc, void* ldst) {
  *(float4*)ldst = *(const float4*)gsrc;        // sync fallback
}
#endif

#if HAVE_ASYNC_COPY
#if __has_builtin(__builtin_amdgcn_s_wait_asynccnt)
#define WAIT_ASYNC(n) __builtin_amdgcn_s_wait_asynccnt(n)
#else
#define WAIT_ASYNC(n) asm volatile("s_wait_asynccnt %0" ::"i"(n) : "memory")
#endif
#define WAIT_DS0() asm volatile("s_wait_dscnt 0x0" ::: "memory")
#else
#define WAIT_ASYNC(n) ((void)0)
#define WAIT_DS0() ((void)0)
#endif

// Copy one 16x64 f32 tile (rows stride EMBED in global) into padded LDS.
// 256 16B chunks, 8 wave-wide steps.
template <int SR>
__device__ __forceinline__ void copy_tile(float (*lds)[SR],
                                          const float* __restrict__ gbase,
                                          int lane) {
#pragma unroll
  for (int it = 0; it < 8; ++it) {
    const int c   = it * 32 + lane;   // 0..255
    const int row = c >> 4;
    const int col = (c & 15) * 4;
    copy16_g2l(gbase + (size_t)row * EMBED + col, &lds[row][col]);
  }
}

// ---------------------------------------------------------------------------
// Phase 1: Q/K/V projections. One wave32 per 64x16 output stripe; the W
// B-fragment is loaded once per k-step and reused by 4 WMMAs.
// Fragment layouts (CDNA5 ISA 7.12.2, 32-bit 16x4 A / 4x16 B):
//   A[m][k]: lane = m + 16*(k>=2), vgpr = k&1
//   B[k][n]: lane = n + 16*(k>=2), vgpr = k&1
//   D[m][n]: vgpr = m&7, lane = n + 16*(m>>3)
// ---------------------------------------------------------------------------
__global__ __launch_bounds__(32) void qkv_gemm_kernel(
    const float* __restrict__ X,
    const float* __restrict__ Wq, const float* __restrict__ Wk,
    const float* __restrict__ Wv,
    float* __restrict__ Q, float* __restrict__ K, float* __restrict__ V) {
  const int lane  = threadIdx.x;
  const int r     = lane & 15;
  const int half  = lane >> 4;
  const int mbase = blockIdx.x * 64;
  const int nbase = blockIdx.y * 16;

  const float* __restrict__ W;
  float* __restrict__ O;
  if (blockIdx.z == 0)      { W = Wq; O = Q; }
  else if (blockIdx.z == 1) { W = Wk; O = K; }
  else                      { W = Wv; O = V; }

  v8f acc[4];
#pragma unroll
  for (int s = 0; s < 4; ++s) acc[s] = (v8f)(0.0f);

#pragma unroll 4
  for (int k0 = 0; k0 < EMBED; k0 += 4) {
    const int k = k0 + 2 * half;
    v2f b;                                       // shared B fragment
    b.x = W[(size_t)k * EMBED + nbase + r];
    b.y = W[(size_t)(k + 1) * EMBED + nbase + r];
#pragma unroll
    for (int s = 0; s < 4; ++s) {
      const float* __restrict__ xr = X + (size_t)(mbase + s * 16 + r) * EMBED;
      v2f a;
      a.x = xr[k];
      a.y = xr[k + 1];
      acc[s] = __builtin_amdgcn_wmma_f32_16x16x4_f32(
          false, a, false, b, (short)0, acc[s], false, false);
    }
  }

#pragma unroll
  for (int s = 0; s < 4; ++s) {
#pragma unroll
    for (int i = 0; i < 8; ++i) {
      const int m = s * 16 + i + 8 * half;
      O[(size_t)(mbase + m) * EMBED + nbase + r] = acc[s][i];
    }
  }
}

// ---------------------------------------------------------------------------
// Phase 2: causal flash attention. One wave32 per (16-query tile, head).
// K/V tiles double-buffered in LDS via async global->LDS copies; fragment
// reads are bank-conflict-free (K rows stride 68, V rows stride 80).
// ---------------------------------------------------------------------------
__global__ __launch_bounds__(32) void flash_attn_kernel(
    const float* __restrict__ Q, const float* __restrict__ K,
    const float* __restrict__ V, float* __restrict__ out) {
  __shared__ __align__(16) float ldsK[2][16][68];
  __shared__ __align__(16) float ldsV[2][16][80];
  __shared__ float pshm[16][17];

  const int lane  = threadIdx.x;
  const int r     = lane & 15;
  const int half  = lane >> 4;
  const int qblk  = blockIdx.x;
  const int head  = blockIdx.y;
  const int qbase = qblk * 16;
  const size_t cbase = (size_t)head * HDIM;
  const float scale = 0.125f;                    // 1/sqrt(64)

  // Kick off async staging of key-tile 0 before anything else.
  copy_tile(ldsK[0], K + cbase, lane);
  copy_tile(ldsV[0], V + cbase, lane);

  // Q A-fragments for all 16 k-steps (16x64 tile, 2 f32 per lane per step)
  v2f qa[16];
  {
    const float* __restrict__ qrow = Q + (size_t)(qbase + r) * EMBED + cbase;
#pragma unroll
    for (int t = 0; t < 16; ++t) {
      const int k = 4 * t + 2 * half;
      qa[t].x = qrow[k];
      qa[t].y = qrow[k + 1];
    }
  }

  v8f o[4];
#pragma unroll
  for (int nc = 0; nc < 4; ++nc) o[nc] = (v8f)(0.0f);
  float rm[8], rs[8];
#pragma unroll
  for (int i = 0; i < 8; ++i) { rm[i] = -INFINITY; rs[i] = 0.0f; }

  for (int kb = 0; kb <= qblk; ++kb) {
    const int cur = kb & 1;

    if (kb < qblk) {
      // Prior reads of the other buffer finished (consumed by WMMA), fence
      // DS anyway, then prefetch tile kb+1 and wait for tile kb (16 older
      // async ops) to land.
      WAIT_DS0();
      const size_t nbase_g = (size_t)(kb + 1) * 16 * EMBED + cbase;
      copy_tile(ldsK[cur ^ 1], K + nbase_g, lane);
      copy_tile(ldsV[cur ^ 1], V + nbase_g, lane);
      WAIT_ASYNC(16);
    } else {
      WAIT_ASYNC(0);
    }
    __syncthreads();

    // ---- S = Q * K^T (K fragments from LDS) ----
    v8f s = (v8f)(0.0f);
#pragma unroll
    for (int t = 0; t < 16; ++t) {
      const int k = 4 * t + 2 * half;
      v2f b;
      b.x = ldsK[cur][r][k];
      b.y = ldsK[cur][r][k + 1];
      s = __builtin_amdgcn_wmma_f32_16x16x4_f32(
          false, qa[t], false, b, (short)0, s, false, false);
    }

    // ---- causal mask + online softmax (row m = i + 8*half, col n = r) ----
    const bool diag = (kb == qblk);
    float p[8];
#pragma unroll
    for (int i = 0; i < 8; ++i) {
      const int m = i + 8 * half;
      float v = s[i] * scale;
      if (diag && (r > m)) v = -INFINITY;

      float mx = v;
      mx = fmaxf(mx, __shfl_xor(mx, 1));
      mx = fmaxf(mx, __shfl_xor(mx, 2));
      mx = fmaxf(mx, __shfl_xor(mx, 4));
      mx = fmaxf(mx, __shfl_xor(mx, 8));

      const float mnew = fmaxf(rm[i], mx);
      const float corr = __expf(rm[i] - mnew);
      const float pv   = __expf(v - mnew);

      float sum = pv;
      sum += __shfl_xor(sum, 1);
      sum += __shfl_xor(sum, 2);
      sum += __shfl_xor(sum, 4);
      sum += __shfl_xor(sum, 8);

      rs[i] = rs[i] * corr + sum;
      rm[i] = mnew;
#pragma unroll
      for (int nc = 0; nc < 4; ++nc) o[nc][i] *= corr;
      p[i] = pv;
    }

    // ---- stage P (D layout) into LDS, re-read in A layout ----
#pragma unroll
    for (int i = 0; i < 8; ++i) pshm[i + 8 * half][r] = p[i];
    __syncthreads();

    // ---- O += P * V (V fragments from LDS) ----
#pragma unroll
    for (int t = 0; t < 4; ++t) {
      const int k = 4 * t + 2 * half;
      v2f a;
      a.x = pshm[r][k];
      a.y = pshm[r][k + 1];
#pragma unroll
      for (int nc = 0; nc < 4; ++nc) {
        v2f b;
        b.x = ldsV[cur][k][nc * 16 + r];
        b.y = ldsV[cur][k + 1][nc * 16 + r];
        o[nc] = __builtin_amdgcn_wmma_f32_16x16x4_f32(
            false, a, false, b, (short)0, o[nc], false, false);
      }
    }
    __syncthreads();
  }

  // ---- epilogue: normalize and store ----
#pragma unroll
  for (int i = 0; i < 8; ++i) {
    const int m = i + 8 * half;
    const float inv = 1.0f / rs[i];
    float* __restrict__ orow = out + (size_t)(qbase + m) * EMBED + cbase;
#pragma unroll
    for (int nc = 0; nc < 4; ++nc) orow[nc * 16 + r] = o[nc][i] * inv;
  }
}

// ---------------------------------------------------------------------------
extern "C" void kernel_launch(void* const* d_in, const int* in_sizes, int n_in,
                              void* d_out, int out_size, void* d_ws, size_t ws_size,
                              hipStream_t stream) {
  (void)in_sizes; (void)n_in; (void)out_size; (void)ws_size;
  const float* x  = (const float*)d_in[0];
  const float* Wq = (const float*)d_in[1];
  const float* Wk = (const float*)d_in[2];
  const float* Wv = (const float*)d_in[3];
  float* out = (float*)d_out;

  float* Q = (float*)d_ws;                       // 3 x 16 MB scratch
  float* K = Q + (size_t)T_DIM * EMBED;
  float* V = K + (size_t)T_DIM * EMBED;

  dim3 g1(T_DIM / 64, EMBED / 16, 3);
  qkv_gemm_kernel<<<g1, dim3(32), 0, stream>>>(x, Wq, Wk, Wv, Q, K, V);

  dim3 g2(T_DIM / 16, HEADS);
  flash_attn_kernel<<<g2, dim3(32), 0, stream>>>(Q, K, V, out);
}